// SameLabelLoss_17214228922603
// MI455X (gfx1250) — compile-verified
//
#include <hip/hip_runtime.h>

typedef __attribute__((ext_vector_type(2))) float v2f;
typedef __attribute__((ext_vector_type(8))) float v8f;

#define B_ROWS   8192
#define DDIM     512
#define NCLS     256
#define NTILES   (B_ROWS / 16)   // 512 tiles of 16 rows

// ---------------- kernel 1: zero class counts ----------------
__global__ __launch_bounds__(256) void k_zero(int* __restrict__ counts) {
    counts[threadIdx.x] = 0;
}

// ---------------- kernel 2: label histogram (integer atomics, exact) -------
__global__ __launch_bounds__(256) void k_histo(const int* __restrict__ labels,
                                               int* __restrict__ counts) {
    int j = blockIdx.x * 256 + threadIdx.x;
    atomicAdd(&counts[labels[j]], 1);
}

// ---------------- kernel 3: WMMA f32 16x16x4 row dot-products --------------
// One wave per 16-row tile. Accumulates:
//   acc_dot = R * Anchor^T  (need diagonal: dot(Z[j], Z[labels[j]]))
//   acc_sq  = R * R^T       (need diagonal: ||Z[j]||^2)
// A-fragment (16x4 f32, 2 VGPRs): lane&15 = M (row), VGPR0 holds K = khalf,
// VGPR1 holds K = khalf+1, where khalf = 2*(lane>>4). B-fragment is the
// transposed layout: lane&15 = N, same K striping -> identical per-lane
// addressing with the anchor row substituted. Passing A as both operands
// yields the symmetric Gram tile.
__global__ __launch_bounds__(256) void k_wmma_dots(const float* __restrict__ Z,
                                                   const int* __restrict__ labels,
                                                   float* __restrict__ rowdot,
                                                   float* __restrict__ rowsq) {
    const int lane  = threadIdx.x & 31;
    const int wave  = threadIdx.x >> 5;
    const int tile  = blockIdx.x * 8 + wave;      // 64 blocks * 8 waves = 512
    const int r0    = tile * 16;
    const int n     = lane & 15;
    const int khalf = (lane >> 4) * 2;            // 0 or 2
    const int myRow = r0 + n;
    const int aRow  = labels[myRow];              // anchor row (< 256, L2-hot)

    const float* pa = Z + (size_t)myRow * DDIM + khalf;
    const float* pb = Z + (size_t)aRow  * DDIM + khalf;

    v8f acc_dot = {};
    v8f acc_sq  = {};

#pragma unroll 4
    for (int k0 = 0; k0 < DDIM; k0 += 4) {
        v2f a = *(const v2f*)(pa + k0);           // global_load_b64, coalesced
        v2f b = *(const v2f*)(pb + k0);
        acc_dot = __builtin_amdgcn_wmma_f32_16x16x4_f32(
            false, a, false, b, (short)0, acc_dot, false, false);
        acc_sq  = __builtin_amdgcn_wmma_f32_16x16x4_f32(
            false, a, false, a, (short)0, acc_sq,  false, false);
    }

    // C/D layout: VGPR v, lanes 0-15 -> (M=v, N=lane); lanes 16-31 -> (M=v+8,
    // N=lane-16). Diagonal M==N lives on lanes 0-7 (v=lane) and lanes 24-31
    // (v=lane-24, M=lane-16).
    const bool valid = (lane < 8) || (lane >= 24);
    const int  v     = (lane < 8) ? lane : (lane - 24);
    const int  m     = (lane < 8) ? lane : (lane - 16);

    float dd = 0.f, ss = 0.f;
    switch (v) {
        case 0: dd = acc_dot[0]; ss = acc_sq[0]; break;
        case 1: dd = acc_dot[1]; ss = acc_sq[1]; break;
        case 2: dd = acc_dot[2]; ss = acc_sq[2]; break;
        case 3: dd = acc_dot[3]; ss = acc_sq[3]; break;
        case 4: dd = acc_dot[4]; ss = acc_sq[4]; break;
        case 5: dd = acc_dot[5]; ss = acc_sq[5]; break;
        case 6: dd = acc_dot[6]; ss = acc_sq[6]; break;
        case 7: dd = acc_dot[7]; ss = acc_sq[7]; break;
    }
    if (valid) {
        rowdot[r0 + m] = dd;
        rowsq [r0 + m] = ss;
    }
}

// ---------------- kernel 4: per-row weighted msd, block tree-reduce --------
__global__ __launch_bounds__(256) void k_combine(const int* __restrict__ labels,
                                                 const int* __restrict__ counts,
                                                 const float* __restrict__ rowdot,
                                                 const float* __restrict__ rowsq,
                                                 float* __restrict__ partials) {
    __shared__ float sm[256];
    const int tid = threadIdx.x;
    const int j   = blockIdx.x * 256 + tid;
    const int c   = labels[j];
    // msd(Z[c], Z[j]) = (||Z[c]||^2 + ||Z[j]||^2 - 2*dot) / D
    float s = (rowsq[c] + rowsq[j] - 2.0f * rowdot[j]) * (1.0f / (float)DDIM);
    float w = (float)(counts[c] - 1);
    sm[tid] = w * s;
    __syncthreads();
    for (int off = 128; off > 0; off >>= 1) {
        if (tid < off) sm[tid] += sm[tid + off];
        __syncthreads();
    }
    if (tid == 0) partials[blockIdx.x] = sm[0];
}

// ---------------- kernel 5: final deterministic reduce + divide ------------
__global__ __launch_bounds__(256) void k_finalize(const int* __restrict__ counts,
                                                  const float* __restrict__ partials,
                                                  float* __restrict__ out) {
    __shared__ float sn[256];
    __shared__ float sd[256];
    const int tid = threadIdx.x;
    sn[tid] = (tid < 32) ? partials[tid] : 0.0f;
    const float nc = (float)counts[tid];          // 256 classes
    sd[tid] = nc * (nc - 1.0f);
    __syncthreads();
    for (int off = 128; off > 0; off >>= 1) {
        if (tid < off) { sn[tid] += sn[tid + off]; sd[tid] += sd[tid + off]; }
        __syncthreads();
    }
    if (tid == 0) out[0] = sn[0] / sd[0];
}

extern "C" void kernel_launch(void* const* d_in, const int* in_sizes, int n_in,
                              void* d_out, int out_size, void* d_ws, size_t ws_size,
                              hipStream_t stream) {
    const float* Z      = (const float*)d_in[0];
    const int*   labels = (const int*)d_in[1];

    float* ws       = (float*)d_ws;
    float* rowdot   = ws;                         // 8192 f32
    float* rowsq    = ws + B_ROWS;                // 8192 f32
    int*   counts   = (int*)(ws + 2 * B_ROWS);    // 256 i32
    float* partials = ws + 2 * B_ROWS + NCLS;     // 32 f32
    float* out      = (float*)d_out;

    k_zero     <<<1,                256, 0, stream>>>(counts);
    k_histo    <<<B_ROWS / 256,     256, 0, stream>>>(labels, counts);
    k_wmma_dots<<<NTILES / 8,       256, 0, stream>>>(Z, labels, rowdot, rowsq);
    k_combine  <<<B_ROWS / 256,     256, 0, stream>>>(labels, counts, rowdot, rowsq, partials);
    k_finalize <<<1,                256, 0, stream>>>(counts, partials, out);
}